// ImprovedVectorizationLoss_73057393705419
// MI455X (gfx1250) — compile-verified
//
#include <hip/hip_runtime.h>
#include <hip/hip_bf16.h>

// Problem constants (from reference setup_inputs)
constexpr int   kB   = 256;
constexpr int   kM   = 4096;
constexpr long long kBM  = (long long)kB * kM;           // 1,048,576 curve entries
constexpr long long kQUADS = kBM / 4;                    // 262,144 4-record quads
constexpr long long kIMG = (long long)kB * 128 * 128;    // 4,194,304 image px
constexpr float kEPS = 1e-7f;
constexpr float kCONT_W  = 0.2f;
constexpr float kRECON_W = 0.1f;

// ws float slots: 0=coordNum 1=widthNum 2=nValid 3=bceSum 4=reconSum 5=contSum
typedef float v2f __attribute__((ext_vector_type(2)));
typedef float v8f __attribute__((ext_vector_type(8)));

// ---- wave32 reduction via V_WMMA_F32_16X16X4_F32 with B = ones (exact in f32)
// A is 16x4: lane m (0-15) holds A[m][0..1] in v0,v1; lane m+16 holds A[m][2..3].
// Partial in v0, zero in v1 -> rowsum[m] = acc[m] + acc[m+16]; D rows replicate
// rowsum across N. Sum of the 8 D VGPRs = half-wave sum (lanes 0-15 vs 16-31);
// one xor-16 shuffle finishes the wave.
__device__ __forceinline__ float wave_reduce_wmma(float acc) {
  v2f a;    a[0] = acc;  a[1] = 0.0f;
  v2f ones; ones[0] = 1.0f; ones[1] = 1.0f;
  v8f c = {};
  c = __builtin_amdgcn_wmma_f32_16x16x4_f32(
      /*neg_a=*/false, a, /*neg_b=*/false, ones,
      /*c_mod=*/(short)0, c, /*reuse_a=*/false, /*reuse_b=*/false);
  float s = c[0] + c[1] + c[2] + c[3] + c[4] + c[5] + c[6] + c[7];
  s += __shfl_xor(s, 16, 32);   // combine the two half-wave sums
  return s;                     // total across all 32 lanes, in every lane
}

__device__ __forceinline__ float wave_reduce_shfl(float v) {
  v += __shfl_xor(v, 1, 32);
  v += __shfl_xor(v, 2, 32);
  v += __shfl_xor(v, 4, 32);
  v += __shfl_xor(v, 8, 32);
  v += __shfl_xor(v, 16, 32);
  return v;
}

__global__ void init_ws_kernel(float* ws) {
  if (threadIdx.x < 8) ws[threadIdx.x] = 0.0f;
}

// ---- masked L1 (coord/width) + n_valid + BCE over B*M 9-float records ------
// 4 records per thread: 144 B = nine 16B-aligned float4 loads per tensor.
__global__ void curve_loss_kernel(const float4* __restrict__ pred4,
                                  const float4* __restrict__ tgt4,
                                  float* __restrict__ ws) {
  float coordAcc = 0.0f, widthAcc = 0.0f, nvAcc = 0.0f, bceAcc = 0.0f;
  const long long stride = (long long)gridDim.x * blockDim.x;
  for (long long q = (long long)blockIdx.x * blockDim.x + threadIdx.x;
       q < kQUADS; q += stride) {
    float4 P[9], T[9];
    #pragma unroll
    for (int k = 0; k < 9; ++k) P[k] = pred4[q * 9 + k];
    #pragma unroll
    for (int k = 0; k < 9; ++k) T[k] = tgt4[q * 9 + k];
    const float* p = (const float*)P;   // 36 floats = 4 records
    const float* t = (const float*)T;
    #pragma unroll
    for (int r = 0; r < 4; ++r) {
      const float* pr = p + r * 9;
      const float* tr = t + r * 9;
      float tv = tr[8];
      float m  = (tv > 0.5f) ? 1.0f : 0.0f;
      float cs = 0.0f;
      #pragma unroll
      for (int c = 0; c < 6; ++c) cs += fabsf(pr[c] - tr[c]);
      float wsum = fabsf(pr[6] - tr[6]) + fabsf(pr[7] - tr[7]);
      coordAcc += m * cs;
      widthAcc += m * wsum;
      nvAcc    += m;
      float pv = fminf(fmaxf(pr[8], kEPS), 1.0f - kEPS);
      bceAcc  += -(tv * __logf(pv) + (1.0f - tv) * __logf(1.0f - pv));
    }
  }
  coordAcc = wave_reduce_shfl(coordAcc);
  widthAcc = wave_reduce_shfl(widthAcc);
  nvAcc    = wave_reduce_shfl(nvAcc);
  bceAcc   = wave_reduce_shfl(bceAcc);

  __shared__ float s[4];
  if (threadIdx.x < 4) s[threadIdx.x] = 0.0f;
  __syncthreads();
  if ((threadIdx.x & 31) == 0) {
    atomicAdd(&s[0], coordAcc);
    atomicAdd(&s[1], widthAcc);
    atomicAdd(&s[2], nvAcc);
    atomicAdd(&s[3], bceAcc);
  }
  __syncthreads();
  if (threadIdx.x == 0) {
    atomicAdd(&ws[0], s[0]);
    atomicAdd(&ws[1], s[1]);
    atomicAdd(&ws[2], s[2]);
    atomicAdd(&ws[3], s[3]);
  }
}

// ---- reconstruction MSE: float4 grid-stride, WMMA wave reduction -----------
__global__ void recon_mse_kernel(const float4* __restrict__ rec,
                                 const float4* __restrict__ img,
                                 float* __restrict__ ws, long long n4) {
  float acc = 0.0f;
  const long long stride = (long long)gridDim.x * blockDim.x;
  for (long long i = (long long)blockIdx.x * blockDim.x + threadIdx.x;
       i < n4; i += stride) {
    float4 r = rec[i];
    float4 g = img[i];
    float dx = r.x - g.x, dy = r.y - g.y, dz = r.z - g.z, dw = r.w - g.w;
    acc = fmaf(dx, dx, acc);
    acc = fmaf(dy, dy, acc);
    acc = fmaf(dz, dz, acc);
    acc = fmaf(dw, dw, acc);
  }
  // whole block alive here -> EXEC all ones, WMMA reduction legal
  float wsum = wave_reduce_wmma(acc);
  __shared__ float s;
  if (threadIdx.x == 0) s = 0.0f;
  __syncthreads();
  if ((threadIdx.x & 31) == 0) atomicAdd(&s, wsum);
  __syncthreads();
  if (threadIdx.x == 0) atomicAdd(&ws[4], s);
}

// ---- continuity: one wave per batch row, ballot-driven pair scan -----------
__global__ void continuity_kernel(const float* __restrict__ pred,
                                  const float* __restrict__ tgt,
                                  float* __restrict__ ws) {
  const int b    = blockIdx.x;     // 256 rows
  const int lane = threadIdx.x;    // 32 lanes (one wave)
  const float* prow = pred + (long long)b * kM * 9;
  const float* trow = tgt  + (long long)b * kM * 9;

  float acc = 0.0f;
  bool  hasCarry = false;
  float cx = 0.0f, cy = 0.0f;      // last valid (x,y) from earlier tiles

  for (int t = 0; t < kM / 32; ++t) {
    int i = t * 32 + lane;
    float tv = trow[(long long)i * 9 + 8];
    bool valid = tv > 0.5f;
    float x = prow[(long long)i * 9 + 4];
    float y = prow[(long long)i * 9 + 5];
    unsigned bal = __builtin_amdgcn_ballot_w32(valid);

    // pair each valid lane with the next valid lane inside this tile
    unsigned above = (lane < 31) ? (bal >> (lane + 1)) : 0u;
    if (valid && above) {
      int j = lane + 1 + __builtin_ctz(above);
      float xj = __shfl(x, j, 32);
      float yj = __shfl(y, j, 32);
      acc += 0.5f * (fabsf(x - xj) + fabsf(y - yj));
    }
    // cross-tile pair: previous tiles' last valid -> this tile's first valid
    if (bal) {
      int f = __builtin_ctz(bal);
      float xf = __shfl(x, f, 32);
      float yf = __shfl(y, f, 32);
      if (hasCarry && lane == 0)
        acc += 0.5f * (fabsf(cx - xf) + fabsf(cy - yf));
      int l = 31 - __builtin_clz(bal);
      cx = __shfl(x, l, 32);
      cy = __shfl(y, l, 32);
      hasCarry = true;
    }
  }
  // full wave alive here -> WMMA reduction legal
  float total = wave_reduce_wmma(acc);
  if (lane == 0) atomicAdd(&ws[5], total);
}

// ---- combine partial sums into the final scalar ----------------------------
__global__ void finalize_kernel(const float* __restrict__ ws,
                                float* __restrict__ out) {
  float coordNum = ws[0], widthNum = ws[1], nv = ws[2];
  float bceSum = ws[3], reconSum = ws[4], contSum = ws[5];
  float coord = (nv > 0.0f) ? coordNum / fmaxf(nv * 6.0f, 1.0f) : 0.0f;
  float width = (nv > 0.0f) ? widthNum / fmaxf(nv * 2.0f, 1.0f) : 0.0f;
  float validity = bceSum / (float)kBM;
  float cont  = contSum / (float)kB;
  float recon = reconSum / (float)kIMG;
  out[0] = coord + width + 2.0f * validity + kCONT_W * cont + kRECON_W * recon;
}

extern "C" void kernel_launch(void* const* d_in, const int* in_sizes, int n_in,
                              void* d_out, int out_size, void* d_ws, size_t ws_size,
                              hipStream_t stream) {
  const float*  pred  = (const float*)d_in[0];   // (256, 4096, 9)
  const float*  tgt   = (const float*)d_in[1];   // (256, 4096, 9)
  const float4* pred4 = (const float4*)d_in[0];
  const float4* tgt4  = (const float4*)d_in[1];
  const float4* rec   = (const float4*)d_in[2];  // (256, 1, 128, 128)
  const float4* img   = (const float4*)d_in[3];
  float* ws  = (float*)d_ws;
  float* out = (float*)d_out;

  init_ws_kernel<<<1, 32, 0, stream>>>(ws);
  curve_loss_kernel<<<1024, 256, 0, stream>>>(pred4, tgt4, ws);
  recon_mse_kernel<<<1024, 256, 0, stream>>>(rec, img, ws, kIMG / 4);
  continuity_kernel<<<kB, 32, 0, stream>>>(pred, tgt, ws);
  finalize_kernel<<<1, 1, 0, stream>>>(ws, out);
}